// GradientNormalizedCrossCorrelation2d_87531433492544
// MI455X (gfx1250) — compile-verified
//
#include <hip/hip_runtime.h>

typedef __attribute__((ext_vector_type(2))) float v2f;
typedef __attribute__((ext_vector_type(8))) float v8f;

// ---------------- workspace layout ----------------
// [0, 32)      : double acc[4]  (per-batch NCC accumulators)
// [256, 256+4MB): float grads[16][256][256]
//                map index = ((img*2 + comp)*4 + b), img in {0,1}, comp in {gx,gy}
#define GRAD_BYTE_OFF 256
#define IMG 256
#define OUTP 248  // 256 - 9 + 1

__global__ void k_init(double* __restrict__ acc) {
    if (threadIdx.x < 4) acc[threadIdx.x] = 0.0;
}

// ---------------- Stage 1: Gaussian blur (reflect) + Sobel (zero pad) ------
// grid = 4(b) * 2(img) * 16(64x64 tiles), block = 256
__global__ __launch_bounds__(256) void k_blur_sobel(const float* __restrict__ x1,
                                                    const float* __restrict__ x2,
                                                    float* __restrict__ grads) {
    __shared__ float in_s[72][73];   // input tile with +/-4 halo (blur 3 + sobel 1)
    __shared__ float tmp_s[72][67];  // horizontally blurred
    __shared__ float blur_s[66][67]; // fully blurred, +/-1 halo for sobel

    const int bid  = blockIdx.x;
    const int tile = bid & 15;
    const int img  = (bid >> 4) & 1;
    const int b    = bid >> 5;
    const int ti   = tile >> 2;
    const int tj   = tile & 3;
    const int tid  = threadIdx.x;

    const float* src = (img == 0 ? x1 : x2) + (size_t)b * (IMG * IMG);

    // normalized 7-tap gaussian, sigma=1 (matches jnp: exp(-0.5*(x/s)^2)/sum)
    float g[7];
    float gs = 0.f;
    #pragma unroll
    for (int k = 0; k < 7; ++k) {
        float d = (float)k - 3.0f;
        g[k] = expf(-0.5f * d * d);
        gs += g[k];
    }
    const float inv_gs = 1.0f / gs;
    #pragma unroll
    for (int k = 0; k < 7; ++k) g[k] *= inv_gs;

    // load 72x72 input tile with reflect indexing (jnp.pad mode='reflect')
    for (int i = tid; i < 72 * 72; i += 256) {
        int r = i / 72, c = i % 72;
        int gi = ti * 64 - 4 + r;
        int gj = tj * 64 - 4 + c;
        gi = gi < 0 ? -gi : (gi > 255 ? 510 - gi : gi);
        gj = gj < 0 ? -gj : (gj > 255 ? 510 - gj : gj);
        in_s[r][c] = src[gi * IMG + gj];
    }
    __syncthreads();

    // horizontal blur: tmp[r][c] covers global cols tj*64-1+c
    for (int i = tid; i < 72 * 66; i += 256) {
        int r = i / 66, c = i % 66;
        float s = 0.f;
        #pragma unroll
        for (int k = 0; k < 7; ++k) s += g[k] * in_s[r][c + k];
        tmp_s[r][c] = s;
    }
    __syncthreads();

    // vertical blur + zero-mask outside image (sobel uses zero padding)
    for (int i = tid; i < 66 * 66; i += 256) {
        int r = i / 66, c = i % 66;
        float s = 0.f;
        #pragma unroll
        for (int k = 0; k < 7; ++k) s += g[k] * tmp_s[r + k][c];
        int gi = ti * 64 - 1 + r;
        int gj = tj * 64 - 1 + c;
        bool inside = (gi >= 0) & (gi < IMG) & (gj >= 0) & (gj < IMG);
        blur_s[r][c] = inside ? s : 0.f;
    }
    __syncthreads();

    float* gx_out = grads + (size_t)((img * 2 + 0) * 4 + b) * (IMG * IMG);
    float* gy_out = grads + (size_t)((img * 2 + 1) * 4 + b) * (IMG * IMG);

    // Sobel (cross-correlation, as XLA conv): Gx=[[1,0,-1],[2,0,-2],[1,0,-1]]
    for (int i = tid; i < 64 * 64; i += 256) {
        int oi = i >> 6, oj = i & 63;
        float a00 = blur_s[oi][oj],     a01 = blur_s[oi][oj + 1],     a02 = blur_s[oi][oj + 2];
        float a10 = blur_s[oi + 1][oj],                               a12 = blur_s[oi + 1][oj + 2];
        float a20 = blur_s[oi + 2][oj], a21 = blur_s[oi + 2][oj + 1], a22 = blur_s[oi + 2][oj + 2];
        float gxv = (a00 - a02) + 2.0f * (a10 - a12) + (a20 - a22);
        float gyv = (a00 + 2.0f * a01 + a02) - (a20 + 2.0f * a21 + a22);
        int gi = ti * 64 + oi, gj = tj * 64 + oj;
        gx_out[gi * IMG + gj] = gxv;
        gy_out[gi * IMG + gj] = gyv;
    }
}

// ---------------- Stage 2: 9x9 box sums via V_WMMA_F32_16X16X4_F32 + NCC ---
// grid = 4(b) * 2(c) * 16 * 16 tiles of 16x16 output positions, block = 32 (1 wave)
__global__ __launch_bounds__(32) void k_ncc(const float* __restrict__ grads,
                                            double* __restrict__ acc) {
    __shared__ float g1_s[24][24];
    __shared__ float g2_s[24][24];
    __shared__ __attribute__((aligned(16))) float V_s[5][16][24]; // S1,S2,S11,S22,S12 vertical 9-sums

    const int bid  = blockIdx.x;
    const int tj   = bid & 15;
    const int ti   = (bid >> 4) & 15;
    const int ch   = (bid >> 8) & 1;
    const int b    = bid >> 9;
    const int lane = threadIdx.x;

    const int j0       = tj * 16;
    const int row_base = ti * 16;

    const float* g1 = grads + (size_t)((0 * 2 + ch) * 4 + b) * (IMG * IMG);
    const float* g2 = grads + (size_t)((1 * 2 + ch) * 4 + b) * (IMG * IMG);

    __builtin_prefetch(g1 + row_base * IMG + j0, 0, 3);
    __builtin_prefetch(g2 + row_base * IMG + j0, 0, 3);

    // load 24x24 gradient subtiles (clamped indices; garbage lands only in masked outputs)
    for (int i = lane; i < 24 * 24; i += 32) {
        int r = i / 24, cc = i % 24;
        int gi = min(row_base + r, IMG - 1);
        int gj = min(j0 + cc, IMG - 1);
        g1_s[r][cc] = g1[gi * IMG + gj];
        g2_s[r][cc] = g2[gi * IMG + gj];
    }
    __syncthreads();

    // vertical 9-row sums of the 5 product maps -> V_s[5][16][24]
    for (int i = lane; i < 16 * 24; i += 32) {
        int r = i / 24, cc = i % 24;
        float s1 = 0.f, s2 = 0.f, s11 = 0.f, s22 = 0.f, s12 = 0.f;
        #pragma unroll
        for (int d = 0; d < 9; ++d) {
            float a = g1_s[r + d][cc];
            float bb = g2_s[r + d][cc];
            s1 += a; s2 += bb;
            s11 = fmaf(a, a, s11);
            s22 = fmaf(bb, bb, s22);
            s12 = fmaf(a, bb, s12);
        }
        V_s[0][r][cc] = s1;  V_s[1][r][cc] = s2;
        V_s[2][r][cc] = s11; V_s[3][r][cc] = s22;
        V_s[4][r][cc] = s12;
    }
    __syncthreads();

    // Horizontal 9-sum as banded GEMM:  C[16x16] = A[16x24] * Band[24x16]
    // A layout (16x4 f32): lane holds row M=lane&15; vgpr j at half h holds k_local=2h+j
    // B layout (4x16 f32): lane holds col N=lane&15; same k_local mapping
    const int n = lane & 15;
    const int h = lane >> 4;

    v8f S[5];
    #pragma unroll
    for (int p = 0; p < 5; ++p) {
        v8f accp = {0.f, 0.f, 0.f, 0.f, 0.f, 0.f, 0.f, 0.f};
        #pragma unroll
        for (int t = 0; t < 6; ++t) {
            const int k0 = 4 * t + 2 * h;                 // k_local 2h, 2h+1 of this K-chunk
            v2f a = *(const v2f*)&V_s[p][n][k0];          // A[m=lane&15][k0..k0+1] (8B aligned)
            v2f bf;
            bf.x = (k0     >= n && k0     <= n + 8) ? 1.0f : 0.0f;  // Band[k][n]=1 iff n<=k<=n+8
            bf.y = (k0 + 1 >= n && k0 + 1 <= n + 8) ? 1.0f : 0.0f;
            accp = __builtin_amdgcn_wmma_f32_16x16x4_f32(
                /*neg_a=*/false, a, /*neg_b=*/false, bf,
                /*c_mod=*/(short)0, accp, /*reuse_a=*/false, /*reuse_b=*/false);
        }
        S[p] = accp;
    }

    // Combine to NCC and reduce. C/D layout: vgpr i -> M=i+8h, N=lane&15.
    const float inv81 = 1.0f / 81.0f;
    float local = 0.f;
    #pragma unroll
    for (int i = 0; i < 8; ++i) {
        int m = i + 8 * h;
        int orow = row_base + m;
        int ocol = j0 + n;
        if (orow < OUTP && ocol < OUTP) {
            float s1 = S[0][i], s2 = S[1][i];
            float s11 = S[2][i], s22 = S[3][i], s12 = S[4][i];
            float mu1 = s1 * inv81, mu2 = s2 * inv81;
            float var1 = s11 * inv81 - mu1 * mu1 + 1e-4f;
            float var2 = s22 * inv81 - mu2 * mu2 + 1e-4f;
            float num = s12 - 81.0f * mu1 * mu2;
            local += num / sqrtf(var1 * var2);
        }
    }
    #pragma unroll
    for (int off = 16; off > 0; off >>= 1)
        local += __shfl_xor(local, off, 32);
    if (lane == 0) atomicAdd(&acc[b], (double)local);
}

__global__ void k_finalize(const double* __restrict__ acc, float* __restrict__ out) {
    int i = threadIdx.x;
    if (i < 4) out[i] = (float)(acc[i] / (2.0 * 81.0 * 248.0 * 248.0));
}

extern "C" void kernel_launch(void* const* d_in, const int* in_sizes, int n_in,
                              void* d_out, int out_size, void* d_ws, size_t ws_size,
                              hipStream_t stream) {
    const float* x1 = (const float*)d_in[0];
    const float* x2 = (const float*)d_in[1];
    float* out = (float*)d_out;

    double* acc  = (double*)d_ws;
    float* grads = (float*)((char*)d_ws + GRAD_BYTE_OFF);

    k_init<<<1, 32, 0, stream>>>(acc);
    k_blur_sobel<<<128, 256, 0, stream>>>(x1, x2, grads);   // 4b * 2img * 16 tiles
    k_ncc<<<2048, 32, 0, stream>>>(grads, acc);             // 4b * 2ch * 256 tiles, 1 wave each
    k_finalize<<<1, 32, 0, stream>>>(acc, out);
}